// CorDBN_Single_44676249813424
// MI455X (gfx1250) — compile-verified
//
#include <hip/hip_runtime.h>
#include <hip/hip_bf16.h>

typedef __attribute__((ext_vector_type(2))) float v2f;
typedef __attribute__((ext_vector_type(8))) float v8f;

#define C_FEAT 64
#define HW     3136            // 56*56
#define B_SZ   128
#define M_TOT  (B_SZ * HW)     // 401408
#define EPS_D  1e-3f
#define EPS_BN 1e-5f

// workspace layout (floats)
#define OFF_SUM  0      // 64   (per-channel sum; sumsq comes free from Gram diagonal)
#define OFF_GRAM 64     // 4096
#define OFF_W2   4160   // 4096
#define OFF_BIAS 8256   // 64
#define WS_ZERO  4160   // zero sums + gram each call

// ---------------------------------------------------------------- init
__global__ __launch_bounds__(256) void init_ws_kernel(float* ws) {
    int i = blockIdx.x * blockDim.x + threadIdx.x;
    for (; i < WS_ZERO; i += gridDim.x * blockDim.x) ws[i] = 0.0f;
}

// ---------------------------------------------------------------- gram + channel sums (fused)
// G = x x^T via f32 WMMA; per-channel Sum accumulated from the same fragments.
// grid: 512 blocks = (b 0..127) x (hw-slice 0..3), 256 threads (8 waves)
__global__ __launch_bounds__(256) void gram_kernel(const float* __restrict__ X,
                                                   float* __restrict__ ws) {
    __shared__ float sG[4096];
    __shared__ float sSum[64];
    const int tid = threadIdx.x;
    for (int i = tid; i < 4096; i += 256) sG[i] = 0.0f;
    if (tid < 64) sSum[tid] = 0.0f;
    __syncthreads();

    const int b     = blockIdx.x >> 2;
    const int slice = blockIdx.x & 3;        // 784 hw each (784 % 4 == 0)
    const int wid  = tid >> 5;
    const int lane = tid & 31;
    const int lm   = lane & 15;
    const int lh   = lane >> 4;
    const float* xb = X + (size_t)b * C_FEAT * HW;

    v8f acc[4][4];
#pragma unroll
    for (int i = 0; i < 4; ++i)
#pragma unroll
        for (int j = 0; j < 4; ++j)
#pragma unroll
            for (int v = 0; v < 8; ++v) acc[i][j][v] = 0.0f;
    float psum[4] = {0.f, 0.f, 0.f, 0.f};

    const int hw_beg = slice * 784;
    const int hw_end = hw_beg + 784;
    for (int hw0 = hw_beg + wid * 4; hw0 < hw_end; hw0 += 8 * 4) {
        v2f frag[4];
#pragma unroll
        for (int g = 0; g < 4; ++g) {
            const float* p = xb + (size_t)(g * 16 + lm) * HW + hw0 + 2 * lh;
            frag[g] = *(const v2f*)p;        // global_load_b64, 8B aligned
        }
#pragma unroll
        for (int g = 0; g < 4; ++g) psum[g] += frag[g][0] + frag[g][1];
#pragma unroll
        for (int i = 0; i < 4; ++i)
#pragma unroll
            for (int j = 0; j < 4; ++j)
                acc[i][j] = __builtin_amdgcn_wmma_f32_16x16x4_f32(
                    false, frag[i], false, frag[j], (short)0, acc[i][j], false, false);
    }

#pragma unroll
    for (int i = 0; i < 4; ++i)
#pragma unroll
        for (int j = 0; j < 4; ++j)
#pragma unroll
            for (int v = 0; v < 8; ++v) {
                int row = i * 16 + v + 8 * lh;
                int col = j * 16 + lm;
                atomicAdd(&sG[row * 64 + col], acc[i][j][v]);  // ds_add_f32
            }
#pragma unroll
    for (int g = 0; g < 4; ++g) atomicAdd(&sSum[g * 16 + lm], psum[g]);
    __syncthreads();
    for (int i = tid; i < 4096; i += 256) atomicAdd(&ws[OFF_GRAM + i], sG[i]);
    if (tid < 64) atomicAdd(&ws[OFF_SUM + tid], sSum[tid]);
}

// ---------------------------------------------------------------- solve: sigma -> sigma^{-1/2} (Newton-Schulz), W2, bias
__global__ __launch_bounds__(256) void solve_kernel(float* __restrict__ ws) {
    __shared__ float sMu[64], sRstd[64];
    __shared__ float sY[4096], sZ[4096], sT[4096];
    __shared__ float sTrace;
    const int tid = threadIdx.x;
    const float inv_m = 1.0f / (float)M_TOT;

    if (tid == 0) sTrace = 0.0f;
    if (tid < 64) {
        float sum = ws[OFF_SUM + tid];
        float sq  = ws[OFF_GRAM + tid * 65];   // Gram diagonal == sum of squares
        float mu  = sum * inv_m;
        float var = (sq - sum * mu) * (1.0f / ((float)M_TOT - 1.0f));
        var = fmaxf(var, 0.0f);
        sMu[tid]   = mu;
        sRstd[tid] = 1.0f / (sqrtf(var) + EPS_BN);
    }
    __syncthreads();

    for (int e = tid; e < 4096; e += 256) {
        int i = e >> 6, j = e & 63;
        float s = sRstd[i] * sRstd[j] * (ws[OFF_GRAM + e] * inv_m - sMu[i] * sMu[j]);
        if (i == j) s += EPS_D;
        sY[e] = s;                       // sigma (temporarily)
        sZ[e] = (i == j) ? 1.0f : 0.0f;  // Z0 = I
    }
    __syncthreads();
    if (tid < 64) atomicAdd(&sTrace, sY[tid * 65]);
    __syncthreads();
    const float tr = sTrace;
    const float invt = 1.0f / tr;
    for (int e = tid; e < 4096; e += 256) sY[e] *= invt;   // Y0 = sigma / tr
    __syncthreads();

    for (int it = 0; it < 25; ++it) {
        // T = 3I - Z*Y
        for (int e = tid; e < 4096; e += 256) {
            int i = e >> 6, j = e & 63;
            float a = 0.0f;
            for (int k = 0; k < 64; ++k) a += sZ[i * 64 + k] * sY[k * 64 + j];
            sT[e] = ((i == j) ? 3.0f : 0.0f) - a;
        }
        __syncthreads();
        // Y = 0.5*Y*T ; Z = 0.5*T*Z (compute into regs, then commit)
        float ny[16], nz[16];
#pragma unroll
        for (int q = 0; q < 16; ++q) {
            int e = tid + q * 256;
            int i = e >> 6, j = e & 63;
            float a = 0.0f, c = 0.0f;
            for (int k = 0; k < 64; ++k) {
                a += sY[i * 64 + k] * sT[k * 64 + j];
                c += sT[i * 64 + k] * sZ[k * 64 + j];
            }
            ny[q] = 0.5f * a; nz[q] = 0.5f * c;
        }
        __syncthreads();
#pragma unroll
        for (int q = 0; q < 16; ++q) {
            int e = tid + q * 256;
            sY[e] = ny[q]; sZ[e] = nz[q];
        }
        __syncthreads();
    }

    // wm = Z / sqrt(tr);  W2[i][j] = wm[i][j]*rstd[j];  bias[i] = sum_j W2[i][j]*mu[j]
    const float rs = rsqrtf(tr);
    for (int e = tid; e < 4096; e += 256) {
        int j = e & 63;
        ws[OFF_W2 + e] = sZ[e] * rs * sRstd[j];
    }
    __syncthreads();
    if (tid < 64) {
        float a = 0.0f;
        for (int j = 0; j < 64; ++j) a += sZ[tid * 64 + j] * rs * sRstd[j] * sMu[j];
        ws[OFF_BIAS + tid] = a;
    }
}

// ---------------------------------------------------------------- apply: out = W2 @ x - bias (transpose folded into addressing)
// grid: 512 blocks = (b 0..127) x (hw-slice 0..3), 256 threads (8 waves)
__global__ __launch_bounds__(256) void apply_kernel(const float* __restrict__ X,
                                                    const float* __restrict__ ws,
                                                    float* __restrict__ out) {
    __shared__ float sW[4096];
    __shared__ float sB[64];
    const int tid = threadIdx.x;
    for (int e = tid; e < 4096; e += 256) sW[e] = ws[OFF_W2 + e];
    if (tid < 64) sB[tid] = ws[OFF_BIAS + tid];
    __syncthreads();

    const int b     = blockIdx.x >> 2;
    const int slice = blockIdx.x & 3;        // 784 hw each (49 chunks of 16)
    const int wid  = tid >> 5;
    const int lane = tid & 31;
    const int lm   = lane & 15;
    const int lh   = lane >> 4;
    const float* xb = X   + (size_t)b * C_FEAT * HW;
    float*       ob = out + (size_t)b * C_FEAT * HW;

    // Hoist all A fragments (W2) out of the m-loop: 4 row-tiles x 16 k-steps
    v2f afrag[4][16];
#pragma unroll
    for (int i = 0; i < 4; ++i)
#pragma unroll
        for (int kk = 0; kk < 16; ++kk) {
            int row = i * 16 + lm;
            int k   = kk * 4 + 2 * lh;
            afrag[i][kk][0] = sW[row * 64 + k];
            afrag[i][kk][1] = sW[row * 64 + k + 1];
        }
    v8f bneg[4];
#pragma unroll
    for (int i = 0; i < 4; ++i)
#pragma unroll
        for (int v = 0; v < 8; ++v) bneg[i][v] = -sB[i * 16 + v + 8 * lh];

    const int hw_beg = slice * 784;
    const int hw_end = hw_beg + 784;
    for (int hw0 = hw_beg + wid * 16; hw0 < hw_end; hw0 += 8 * 16) {
        v8f acc[4];
#pragma unroll
        for (int i = 0; i < 4; ++i) acc[i] = bneg[i];
#pragma unroll
        for (int kk = 0; kk < 16; ++kk) {
            int k = kk * 4 + 2 * lh;
            v2f bf;
            bf[0] = xb[(size_t)k       * HW + hw0 + lm];  // 64B coalesced per half-wave
            bf[1] = xb[(size_t)(k + 1) * HW + hw0 + lm];
#pragma unroll
            for (int i = 0; i < 4; ++i)
                acc[i] = __builtin_amdgcn_wmma_f32_16x16x4_f32(
                    false, afrag[i][kk], false, bf, (short)0, acc[i], false, false);
        }
#pragma unroll
        for (int i = 0; i < 4; ++i)
#pragma unroll
            for (int v = 0; v < 8; ++v)
                __builtin_nontemporal_store(acc[i][v],
                    &ob[(size_t)(i * 16 + v + 8 * lh) * HW + hw0 + lm]);
    }
}

// ---------------------------------------------------------------- launch
extern "C" void kernel_launch(void* const* d_in, const int* in_sizes, int n_in,
                              void* d_out, int out_size, void* d_ws, size_t ws_size,
                              hipStream_t stream) {
    (void)in_sizes; (void)n_in; (void)out_size; (void)ws_size;
    const float* X  = (const float*)d_in[0];
    float*       ws = (float*)d_ws;
    float*       Y  = (float*)d_out;

    init_ws_kernel<<<17, 256, 0, stream>>>(ws);
    gram_kernel   <<<512, 256, 0, stream>>>(X, ws);
    solve_kernel  <<<1, 256, 0, stream>>>(ws);
    apply_kernel  <<<512, 256, 0, stream>>>(X, ws, Y);
}